// RepCONC_75110388073017
// MI455X (gfx1250) — compile-verified
//
#include <hip/hip_runtime.h>
#include <hip/hip_bf16.h>

typedef __attribute__((ext_vector_type(2))) float v2f;
typedef __attribute__((ext_vector_type(8))) float v8f;

#define B_N 4096
#define H_N 768
#define M_N 48
#define K_N 256
#define D_N 16
#define BT  128
#define NBT (B_N / BT)   // 32 b-tiles

// ---- order-preserving float<->uint key (for deterministic atomic min/max) ----
__device__ __forceinline__ unsigned f32_key(float f) {
  unsigned u = __float_as_uint(f);
  return (u & 0x80000000u) ? ~u : (u | 0x80000000u);
}
__device__ __forceinline__ float key_f32(unsigned k) {
  return __uint_as_float((k & 0x80000000u) ? (k & 0x7FFFFFFFu) : ~k);
}
__device__ __forceinline__ v2f mkv2(float x, float y) { v2f t; t[0] = x; t[1] = y; return t; }

// =====================================================================
// Kernel 1: rotated = dense @ rotation^T   (4096x768 * 768x768, fp32 WMMA)
// Block = 256 thr (8 waves), tile 128(M) x 128(N), waves 4(M)x2(N),
// 32x64 per wave = 2x4 register tiles.
// LDS holds K-PAIRS as float2 so each WMMA fragment is ONE aligned
// ds_load_b64 into an even VGPR pair (no packing movs).
// =====================================================================
__global__ __launch_bounds__(256)
void rot_gemm_kernel(const float* __restrict__ A, const float* __restrict__ Rw,
                     float* __restrict__ Out) {
  __shared__ v2f As[16][132];   // As[kp][m] = {A[m][2kp], A[m][2kp+1]}
  __shared__ v2f Bs[16][132];   // Bs[kp][n] = {Rw[n][2kp], Rw[n][2kp+1]}

  const int tid  = threadIdx.x;
  const int wv   = tid >> 5;
  const int L    = tid & 31;
  const int lm   = L & 15;
  const int kph  = (L >> 4);      // K half-select: pair index offset 0 or 1
  const int half8 = (L >> 4) * 8; // C/D M split at half-wave per ISA layout
  const int b0 = blockIdx.x * 128;
  const int n0 = blockIdx.y * 128;
  const int mw = (wv >> 1) * 32;
  const int nw = (wv & 1) * 64;

  const v8f vz = {};
  v8f acc[2][4];
  #pragma unroll
  for (int i = 0; i < 2; ++i)
    #pragma unroll
    for (int j = 0; j < 4; ++j) acc[i][j] = vz;

  for (int ko = 0; ko < H_N; ko += 32) {
    // A tile 128x32 -> pair-transposed
    #pragma unroll
    for (int it = 0; it < 4; ++it) {
      int idx = tid + it * 256;
      int r = idx >> 3, q = (idx & 7) * 4;
      const float4 v = *(const float4*)(A + (size_t)(b0 + r) * H_N + ko + q);
      As[(q >> 1) + 0][r] = mkv2(v.x, v.y);
      As[(q >> 1) + 1][r] = mkv2(v.z, v.w);
    }
    // B tile 128x32: Bs[kp][n] from Rw rows (coalesced), pair-transposed
    #pragma unroll
    for (int it = 0; it < 4; ++it) {
      int idx = tid + it * 256;
      int r = idx >> 3, q = (idx & 7) * 4;
      const float4 v = *(const float4*)(Rw + (size_t)(n0 + r) * H_N + ko + q);
      Bs[(q >> 1) + 0][r] = mkv2(v.x, v.y);
      Bs[(q >> 1) + 1][r] = mkv2(v.z, v.w);
    }
    __syncthreads();
    #pragma unroll
    for (int kc = 0; kc < 32; kc += 4) {
      const int kp = (kc >> 1) + kph;          // lane's K-pair row
      v2f a[2], f[4];
      #pragma unroll
      for (int i = 0; i < 2; ++i) a[i] = As[kp][mw + i * 16 + lm];
      #pragma unroll
      for (int j = 0; j < 4; ++j) f[j] = Bs[kp][nw + j * 16 + lm];
      #pragma unroll
      for (int i = 0; i < 2; ++i)
        #pragma unroll
        for (int j = 0; j < 4; ++j)
          acc[i][j] = __builtin_amdgcn_wmma_f32_16x16x4_f32(false, a[i], false, f[j],
                                                            (short)0, acc[i][j], false, false);
    }
    __syncthreads();
  }
  #pragma unroll
  for (int i = 0; i < 2; ++i)
    #pragma unroll
    for (int j = 0; j < 4; ++j)
      #pragma unroll
      for (int e = 0; e < 8; ++e) {
        int row = b0 + mw + i * 16 + e + half8;
        int col = n0 + nw + j * 16 + lm;
        Out[(size_t)row * H_N + col] = acc[i][j][e];
      }
}

// =====================================================================
// Distance-tile passes. Each block: m = blockIdx.y, 128 b-rows, all 256 k.
// cross = x(128x16) @ cent^T(16x256) via 4 chained f32 WMMAs per 16x16 tile;
// x A-fragments hoisted to registers, centroid B-frags are single b64 loads.
// MODE 0: global min/max of distances (ordered-uint atomics, order-free)
// MODE 1: u_k partial = sum_b exp((mid-D)/amp)*C_b   -> per-tile scratch
// MODE 2: v_b = sum_k E*R_k;  C_b = 1/(B*v_b)
// MODE 3: argmax_k (logR_k - D/amp)  -> codes + decode (gather centroid row)
// =====================================================================
template<int MODE>
__global__ __launch_bounds__(256)
void dist_pass_kernel(const float* __restrict__ rotated,
                      const float* __restrict__ centroids,
                      const float* __restrict__ stats,
                      const float* __restrict__ Cprev,
                      const float* __restrict__ u_in,
                      float* __restrict__ part_u,
                      float* __restrict__ C_out,
                      unsigned* __restrict__ mmax,
                      unsigned* __restrict__ mmin,
                      float* __restrict__ out_quant,
                      float* __restrict__ out_codes) {
  __shared__ v2f   cTp[8][260];     // cTp[dp][k] = {cent[k][2dp], cent[k][2dp+1]}
  __shared__ float xs[128][18];     // x tile; pad 18 keeps (row, even-d) 8B aligned
  __shared__ float xsq[128];        // |x|^2 per row
  __shared__ float c2s[256];        // |c|^2 per k
  __shared__ float sk[256];         // R / logR / wave-reduce scratch
  __shared__ float spart[256][17];  // deterministic per-column partials (MODE 1)

  const int tid  = threadIdx.x;
  const int m    = blockIdx.y;
  const int b0   = blockIdx.x * BT;
  const int wv   = tid >> 5;
  const int L    = tid & 31;
  const int lm   = L & 15;
  const int kph  = (L >> 4);        // K-pair half select
  const int half8 = (L >> 4) * 8;

  #pragma unroll
  for (int it = 0; it < 4; ++it) {              // centroids -> pair-transposed
    int idx = tid + it * 256;
    int k = idx >> 2, q = (idx & 3) * 4;
    const float4 v = *(const float4*)(centroids + ((size_t)m * K_N + k) * D_N + q);
    cTp[(q >> 1) + 0][k] = mkv2(v.x, v.y);
    cTp[(q >> 1) + 1][k] = mkv2(v.z, v.w);
  }
  #pragma unroll
  for (int it = 0; it < 2; ++it) {              // x tile
    int idx = tid + it * 256;
    int r = idx >> 2, q = (idx & 3) * 4;
    const float4 v = *(const float4*)(rotated + (size_t)(b0 + r) * H_N + m * D_N + q);
    xs[r][q+0] = v.x; xs[r][q+1] = v.y; xs[r][q+2] = v.z; xs[r][q+3] = v.w;
  }
  __syncthreads();
  if (tid < 128) {
    float s = 0.f;
    #pragma unroll
    for (int d = 0; d < D_N; ++d) { float x = xs[tid][d]; s += x * x; }
    xsq[tid] = s;
  }
  {
    float s = 0.f;
    #pragma unroll
    for (int dp = 0; dp < 8; ++dp) {
      v2f c = cTp[dp][tid];
      s += c[0] * c[0] + c[1] * c[1];
    }
    c2s[tid] = s;
  }
  if constexpr (MODE == 2) sk[tid] = 1.0f / (256.0f * u_in[(size_t)m * K_N + tid]);
  if constexpr (MODE == 3) sk[tid] = -__logf(256.0f * u_in[(size_t)m * K_N + tid]);
  __syncthreads();

  float mid = 0.f, inva = 0.f;
  if constexpr (MODE != 0) { mid = stats[2*m]; inva = 1.0f / stats[2*m+1]; }

  float xq[8];
  #pragma unroll
  for (int j = 0; j < 8; ++j) xq[j] = xsq[wv * 16 + j + half8];

  // Hoist A-fragments (invariant across k-tiles): single aligned b64 loads.
  v2f af[4];
  #pragma unroll
  for (int d0 = 0; d0 < 16; d0 += 4)
    af[d0 >> 2] = *(const v2f*)&xs[wv * 16 + lm][d0 + kph * 2];

  float cb[8];
  if constexpr (MODE == 1) {
    #pragma unroll
    for (int j = 0; j < 8; ++j) {
      int b = b0 + wv * 16 + j + half8;
      cb[j] = Cprev ? Cprev[(size_t)m * B_N + b] : 1.0f;
    }
  }

  float mn = 3.402823466e38f, mx = -3.402823466e38f;
  float av[8], bs[8]; int bk[8];
  #pragma unroll
  for (int j = 0; j < 8; ++j) { av[j] = 0.f; bs[j] = -3.402823466e38f; bk[j] = 0; }
  const v8f vz = {};

  for (int t = 0; t < 16; ++t) {                // 16 k-tiles of 16
    const int kc0 = t * 16;
    v8f acc = vz;
    #pragma unroll
    for (int d0 = 0; d0 < 16; d0 += 4) {
      v2f f = cTp[(d0 >> 1) + kph][kc0 + lm];
      acc = __builtin_amdgcn_wmma_f32_16x16x4_f32(false, af[d0 >> 2], false, f,
                                                  (short)0, acc, false, false);
    }
    const float c2v = c2s[kc0 + lm];
    if constexpr (MODE == 0) {
      #pragma unroll
      for (int j = 0; j < 8; ++j) {
        float dist = xq[j] + c2v - 2.0f * acc[j];
        mn = fminf(mn, dist); mx = fmaxf(mx, dist);
      }
    } else if constexpr (MODE == 1) {
      float lu = 0.f;
      #pragma unroll
      for (int j = 0; j < 8; ++j) {
        float dist = xq[j] + c2v - 2.0f * acc[j];
        lu += __expf((mid - dist) * inva) * cb[j];
      }
      spart[kc0 + lm][wv * 2 + (L >> 4)] = lu;   // each slot written exactly once
    } else if constexpr (MODE == 2) {
      float rk = sk[kc0 + lm];
      #pragma unroll
      for (int j = 0; j < 8; ++j) {
        float dist = xq[j] + c2v - 2.0f * acc[j];
        av[j] += __expf((mid - dist) * inva) * rk;
      }
    } else {
      float lr = sk[kc0 + lm];
      #pragma unroll
      for (int j = 0; j < 8; ++j) {
        float dist = xq[j] + c2v - 2.0f * acc[j];
        float s = lr - dist * inva;              // mid/amp constants drop out
        if (s > bs[j]) { bs[j] = s; bk[j] = kc0 + lm; }
      }
    }
  }

  if constexpr (MODE == 0) {
    #pragma unroll
    for (int msk = 16; msk >= 1; msk >>= 1) {
      mn = fminf(mn, __shfl_xor(mn, msk, 32));
      mx = fmaxf(mx, __shfl_xor(mx, msk, 32));
    }
    if (L == 0) { sk[wv] = mn; sk[8 + wv] = mx; }
    __syncthreads();
    if (tid == 0) {
      float amn = sk[0], amx = sk[8];
      #pragma unroll
      for (int w = 1; w < 8; ++w) { amn = fminf(amn, sk[w]); amx = fmaxf(amx, sk[8 + w]); }
      atomicMin(&mmin[m], f32_key(amn));
      atomicMax(&mmax[m], f32_key(amx));
    }
  } else if constexpr (MODE == 1) {
    __syncthreads();
    float s = 0.f;
    #pragma unroll
    for (int q = 0; q < 16; ++q) s += spart[tid][q];   // fixed order: deterministic
    part_u[((size_t)m * NBT + blockIdx.x) * K_N + tid] = s;
  } else if constexpr (MODE == 2) {
    #pragma unroll
    for (int j = 0; j < 8; ++j) {
      #pragma unroll
      for (int msk = 1; msk <= 8; msk <<= 1)
        av[j] += __shfl_xor(av[j], msk, 32);           // sum over 16 k-lanes
    }
    if (lm == 0) {
      #pragma unroll
      for (int j = 0; j < 8; ++j) {
        int b = b0 + wv * 16 + j + half8;
        C_out[(size_t)m * B_N + b] = 1.0f / (4096.0f * av[j]);
      }
    }
  } else {
    #pragma unroll
    for (int j = 0; j < 8; ++j) {
      #pragma unroll
      for (int msk = 1; msk <= 8; msk <<= 1) {
        float os = __shfl_xor(bs[j], msk, 32);
        int   ok = __shfl_xor(bk[j], msk, 32);
        if (os > bs[j] || (os == bs[j] && ok < bk[j])) { bs[j] = os; bk[j] = ok; }
      }
    }
    if (lm == 0) {
      #pragma unroll
      for (int j = 0; j < 8; ++j) {
        int b = b0 + wv * 16 + j + half8;
        int code = bk[j];
        out_codes[(size_t)b * M_N + m] = (float)code;
        const float4* src = (const float4*)(centroids + ((size_t)m * K_N + code) * D_N);
        float4* dst = (float4*)(out_quant + (size_t)b * H_N + m * D_N);
        dst[0] = src[0]; dst[1] = src[1]; dst[2] = src[2]; dst[3] = src[3];
      }
    }
  }
}

// ---------------- small helper kernels ----------------
__global__ void init_ws_kernel(unsigned* mmax, unsigned* mmin) {
  int i = threadIdx.x;
  if (i < M_N) { mmax[i] = 0u; mmin[i] = 0xFFFFFFFFu; }
}
__global__ void stats_kernel(const unsigned* __restrict__ mmax,
                             const unsigned* __restrict__ mmin,
                             float* __restrict__ stats) {
  int i = threadIdx.x;
  if (i < M_N) {
    float mx = key_f32(mmax[i]);
    float mn = key_f32(mmin[i]);
    float mid = 0.5f * (mx + mn);
    stats[2*i]   = mid;
    stats[2*i+1] = mx - mid + 1e-5f;
  }
}
__global__ void reduce_u_kernel(const float* __restrict__ part, float* __restrict__ u) {
  int i = blockIdx.x * 256 + threadIdx.x;   // i = m*256 + k
  int m2 = i >> 8, k = i & 255;
  float s = 0.f;
  for (int t = 0; t < NBT; ++t) s += part[((size_t)m2 * NBT + t) * K_N + k];
  u[i] = s;
}

// =====================================================================
extern "C" void kernel_launch(void* const* d_in, const int* in_sizes, int n_in,
                              void* d_out, int out_size, void* d_ws, size_t ws_size,
                              hipStream_t stream) {
  (void)in_sizes; (void)n_in; (void)out_size; (void)ws_size;
  const float* dense = (const float*)d_in[0];   // 4096x768
  const float* rot   = (const float*)d_in[1];   // 768x768
  const float* cent  = (const float*)d_in[2];   // 48x256x16

  float* out     = (float*)d_out;
  float* rotated = out;                               // 4096*768
  float* quant   = out + (size_t)B_N * H_N;           // 4096*768
  float* codes   = quant + (size_t)B_N * H_N;         // 4096*48 (as float)

  float* ws    = (float*)d_ws;
  float* part  = ws;                                  // 48*32*256
  float* u1    = part + (size_t)M_N * NBT * K_N;      // 48*256
  float* u2    = u1 + M_N * K_N;
  float* u3    = u2 + M_N * K_N;
  float* C1    = u3 + M_N * K_N;                      // 48*4096
  float* C2    = C1 + (size_t)M_N * B_N;
  float* stats = C2 + (size_t)M_N * B_N;              // 96
  unsigned* mmax = (unsigned*)(stats + 2 * M_N);      // 48
  unsigned* mmin = mmax + M_N;                        // 48

  dim3 blk(256);
  dim3 dg(NBT, M_N);

  init_ws_kernel<<<1, 64, 0, stream>>>(mmax, mmin);
  rot_gemm_kernel<<<dim3(B_N / 128, H_N / 128), blk, 0, stream>>>(dense, rot, rotated);

  // pass 1: distance min/max per m
  dist_pass_kernel<0><<<dg, blk, 0, stream>>>(rotated, cent, nullptr, nullptr, nullptr,
                                              nullptr, nullptr, mmax, mmin, nullptr, nullptr);
  stats_kernel<<<1, 64, 0, stream>>>(mmax, mmin, stats);

  // sinkhorn iter 1
  dist_pass_kernel<1><<<dg, blk, 0, stream>>>(rotated, cent, stats, nullptr, nullptr,
                                              part, nullptr, nullptr, nullptr, nullptr, nullptr);
  reduce_u_kernel<<<M_N, 256, 0, stream>>>(part, u1);
  dist_pass_kernel<2><<<dg, blk, 0, stream>>>(rotated, cent, stats, nullptr, u1,
                                              nullptr, C1, nullptr, nullptr, nullptr, nullptr);
  // iter 2
  dist_pass_kernel<1><<<dg, blk, 0, stream>>>(rotated, cent, stats, C1, nullptr,
                                              part, nullptr, nullptr, nullptr, nullptr, nullptr);
  reduce_u_kernel<<<M_N, 256, 0, stream>>>(part, u2);
  dist_pass_kernel<2><<<dg, blk, 0, stream>>>(rotated, cent, stats, nullptr, u2,
                                              nullptr, C2, nullptr, nullptr, nullptr, nullptr);
  // iter 3 (only the column (per-k) factors are needed for argmax)
  dist_pass_kernel<1><<<dg, blk, 0, stream>>>(rotated, cent, stats, C2, nullptr,
                                              part, nullptr, nullptr, nullptr, nullptr, nullptr);
  reduce_u_kernel<<<M_N, 256, 0, stream>>>(part, u3);

  // argmax + decode
  dist_pass_kernel<3><<<dg, blk, 0, stream>>>(rotated, cent, stats, nullptr, u3,
                                              nullptr, nullptr, nullptr, nullptr, quant, codes);
}